// VitVQModule_7988639171037
// MI455X (gfx1250) — compile-verified
//
#include <hip/hip_runtime.h>

typedef __attribute__((ext_vector_type(2))) float v2f;
typedef __attribute__((ext_vector_type(4))) float v4f;
typedef __attribute__((ext_vector_type(8))) float v8f;
typedef __attribute__((ext_vector_type(4))) int   v4i;

#define E_DIM 32
#define N_TOK 32768   // 32*1024
#define N_EMB 8192
#define KC 128        // codes staged in LDS per pass
#define LDSW 36       // dword stride per code row in LDS (conflict-free b64, 16B-aligned stores)
#define NSTAGE (N_EMB / KC)
#define EPS 1e-12f

#if defined(__gfx1250__) && __has_builtin(__builtin_amdgcn_global_load_async_to_lds_b128)
#define USE_ASYNC 1
#else
#define USE_ASYNC 0
#endif

// ---------------- Kernel 1/2: L2 row normalize (rows of 32 floats) ----------------
__global__ __launch_bounds__(256) void row_l2norm_kernel(const float* __restrict__ src,
                                                         float* __restrict__ dst, int nrows) {
    int r = blockIdx.x * blockDim.x + threadIdx.x;
    if (r >= nrows) return;
    const float* p = src + (size_t)r * E_DIM;
    v4f v[8];
    float ss = 0.f;
    #pragma unroll
    for (int i = 0; i < 8; ++i) {
        v[i] = *(const v4f*)(p + 4 * i);
        ss += v[i].x * v[i].x + v[i].y * v[i].y + v[i].z * v[i].z + v[i].w * v[i].w;
    }
    float s = rsqrtf(ss + EPS);
    float* q = dst + (size_t)r * E_DIM;
    #pragma unroll
    for (int i = 0; i < 8; ++i) {
        v4f o = { v[i].x * s, v[i].y * s, v[i].z * s, v[i].w * s };
        *(v4f*)(q + 4 * i) = o;
    }
}

// ---------------- Kernel 3: WMMA f32 nearest-code search (argmax of dot) ----------------
// Block = 256 threads (8 waves). Each wave owns 32 tokens (two 16-row A tiles sharing B).
// Codebook double-buffered through LDS; async global->LDS copies overlap the WMMA chains.
__global__ __launch_bounds__(256) void vq_search_kernel(const float* __restrict__ flat,   // N_TOK x 32, normalized
                                                        const float* __restrict__ emb,    // N_EMB x 32, normalized
                                                        int* __restrict__ best_idx,
                                                        float* __restrict__ idx_f_out) {
    __shared__ float lds[2][KC * LDSW];   // 2 x 18432 B

    const int tid  = threadIdx.x;
    const int lane = tid & 31;
    const int wave = tid >> 5;
    const bool hi  = lane >= 16;
    const int  l16 = lane & 15;
    const int  m0  = blockIdx.x * 256 + wave * 32;   // first token row of this wave

    // A fragments for both 16-row tiles: chunk c covers dot-dims 4c..4c+3.
    // lanes 0-15 hold {4c,4c+1}, lanes 16-31 hold {4c+2,4c+3} -> one aligned float2 each.
    v2f a0[8], a1[8];
    {
        const float* ar0 = flat + (size_t)(m0 + l16) * E_DIM;
        const float* ar1 = flat + (size_t)(m0 + 16 + l16) * E_DIM;
        #pragma unroll
        for (int c = 0; c < 8; ++c) {
            a0[c] = *(const v2f*)(ar0 + 4 * c + (hi ? 2 : 0));
            a1[c] = *(const v2f*)(ar1 + 4 * c + (hi ? 2 : 0));
        }
    }

    float bv0[8], bv1[8];
    int   bi0[8], bi1[8];
    #pragma unroll
    for (int j = 0; j < 8; ++j) {
        bv0[j] = -3.4e38f; bi0[j] = 0;
        bv1[j] = -3.4e38f; bi1[j] = 0;
    }

    // ---- stage issue: 128 codes x 32 floats -> padded LDS buffer ----
    // thread does 4 x b128 chunks; lds byte off = r*144 + c4*16 (16B aligned).
#if USE_ASYNC
    #define STAGE_ISSUE(buf, gbase)                                                          \
        {   const float* _g = (gbase);                                                       \
            _Pragma("unroll")                                                                \
            for (int _i = 0; _i < 4; ++_i) {                                                 \
                int _t = tid + _i * 256;                                                     \
                int _r = _t >> 3, _c4 = _t & 7;                                              \
                __builtin_amdgcn_global_load_async_to_lds_b128(                              \
                    (__attribute__((address_space(1))) v4i*)(float*)(_g + (size_t)_r * E_DIM + 4 * _c4), \
                    (__attribute__((address_space(3))) v4i*)&(buf)[_r * LDSW + 4 * _c4],     \
                    0, 0);                                                                   \
            }                                                                                \
        }
    #define STAGE_DRAIN() asm volatile("s_wait_asynccnt 0x0" ::: "memory")
#else
    #define STAGE_ISSUE(buf, gbase)                                                          \
        {   const float* _g = (gbase);                                                       \
            _Pragma("unroll")                                                                \
            for (int _i = 0; _i < 4; ++_i) {                                                 \
                int _t = tid + _i * 256;                                                     \
                int _r = _t >> 3, _c4 = _t & 7;                                              \
                v4f _v = *(const v4f*)(_g + (size_t)_r * E_DIM + 4 * _c4);                   \
                *(v4f*)&(buf)[_r * LDSW + 4 * _c4] = _v;                                     \
            }                                                                                \
        }
    #define STAGE_DRAIN() ((void)0)
#endif

    // prologue: fill buffer 0
    STAGE_ISSUE(lds[0], emb);
    STAGE_DRAIN();
    __syncthreads();

    for (int s = 0; s < NSTAGE; ++s) {
        const float* cur = lds[s & 1];
        if (s + 1 < NSTAGE) {
            STAGE_ISSUE(lds[(s + 1) & 1], emb + (size_t)(s + 1) * KC * E_DIM);
        }

        #pragma unroll 1
        for (int kt = 0; kt < KC / 16; ++kt) {     // 8 N-tiles of 16 codes
            const int cl = kt * 16 + l16;          // code (local) this lane supplies for B
            v2f b[8];
            #pragma unroll
            for (int c = 0; c < 8; ++c)            // ds_load_b64, conflict-free (stride 36 dw)
                b[c] = *(const v2f*)&cur[cl * LDSW + 4 * c + (hi ? 2 : 0)];

            v8f acc0 = {}, acc1 = {};
            #pragma unroll
            for (int c = 0; c < 8; ++c) {          // two independent chains -> ILP
                acc0 = __builtin_amdgcn_wmma_f32_16x16x4_f32(
                        false, a0[c], false, b[c], (short)0, acc0, false, false);
                acc1 = __builtin_amdgcn_wmma_f32_16x16x4_f32(
                        false, a1[c], false, b[c], (short)0, acc1, false, false);
            }

            // D layout: lane column N = lane&15, VGPR j -> row (tile_base + j + (hi?8:0)).
            const int kcol = s * KC + kt * 16 + l16;
            #pragma unroll
            for (int j = 0; j < 8; ++j) {
                float x0 = acc0[j];
                if (x0 > bv0[j]) { bv0[j] = x0; bi0[j] = kcol; }  // strict > keeps lowest k on ties
                float x1 = acc1[j];
                if (x1 > bv1[j]) { bv1[j] = x1; bi1[j] = kcol; }
            }
        }

        STAGE_DRAIN();
        __syncthreads();
    }

    // Cross-lane argmax within each 16-lane half (xor 8,4,2,1 stays inside the half).
    #pragma unroll
    for (int j = 0; j < 8; ++j) {
        float v0 = bv0[j]; int i0 = bi0[j];
        float v1 = bv1[j]; int i1 = bi1[j];
        #pragma unroll
        for (int off = 8; off > 0; off >>= 1) {
            float ov = __shfl_xor(v0, off, 32);
            int   oi = __shfl_xor(i0, off, 32);
            if (ov > v0 || (ov == v0 && oi < i0)) { v0 = ov; i0 = oi; }
            ov = __shfl_xor(v1, off, 32);
            oi = __shfl_xor(i1, off, 32);
            if (ov > v1 || (ov == v1 && oi < i1)) { v1 = ov; i1 = oi; }
        }
        if (l16 == 0) {
            int r0 = m0 + j + (hi ? 8 : 0);
            int r1 = r0 + 16;
            best_idx[r0]  = i0;
            idx_f_out[r0] = (float)i0;
            best_idx[r1]  = i1;
            idx_f_out[r1] = (float)i1;
        }
    }
}

// ---------------- Kernel 4: gather + renormalize z_q, partial loss sums ----------------
__global__ __launch_bounds__(256) void gather_loss_kernel(const float* __restrict__ emb_raw,
                                                          const float* __restrict__ flat_norm,
                                                          const int* __restrict__ best_idx,
                                                          float* __restrict__ zq_out,
                                                          float* __restrict__ partials) {
    __shared__ float red[256];
    int tok = blockIdx.x * blockDim.x + threadIdx.x;
    float lsum = 0.f;
    if (tok < N_TOK) {
        int idx = best_idx[tok];
        const float* e = emb_raw + (size_t)idx * E_DIM;
        v4f v[8];
        float ss = 0.f;
        #pragma unroll
        for (int i = 0; i < 8; ++i) {
            v[i] = *(const v4f*)(e + 4 * i);
            ss += v[i].x * v[i].x + v[i].y * v[i].y + v[i].z * v[i].z + v[i].w * v[i].w;
        }
        float sc = rsqrtf(ss + EPS);
        const float* h = flat_norm + (size_t)tok * E_DIM;
        float* q = zq_out + (size_t)tok * E_DIM;
        #pragma unroll
        for (int i = 0; i < 8; ++i) {
            v4f hn = *(const v4f*)(h + 4 * i);
            v4f z  = { v[i].x * sc, v[i].y * sc, v[i].z * sc, v[i].w * sc };
            *(v4f*)(q + 4 * i) = z;
            float dx = z.x - hn.x, dy = z.y - hn.y, dz = z.z - hn.z, dw = z.w - hn.w;
            lsum += dx * dx + dy * dy + dz * dz + dw * dw;
        }
    }
    red[threadIdx.x] = lsum;
    __syncthreads();
    for (int st = 128; st > 0; st >>= 1) {       // deterministic tree
        if (threadIdx.x < st) red[threadIdx.x] += red[threadIdx.x + st];
        __syncthreads();
    }
    if (threadIdx.x == 0) partials[blockIdx.x] = red[0];
}

// ---------------- Kernel 5: fixed-order final loss reduction ----------------
__global__ void finalize_loss_kernel(const float* __restrict__ partials, int n,
                                     float* __restrict__ loss_out) {
    if (threadIdx.x == 0 && blockIdx.x == 0) {
        float s = 0.f;
        for (int i = 0; i < n; ++i) s += partials[i];
        float m = s / (float)(N_TOK * E_DIM);
        loss_out[0] = m;   // q_latent_loss (numerically identical to e_latent_loss)
        loss_out[1] = m;   // e_latent_loss
    }
}

extern "C" void kernel_launch(void* const* d_in, const int* in_sizes, int n_in,
                              void* d_out, int out_size, void* d_ws, size_t ws_size,
                              hipStream_t stream) {
    const float* hidden = (const float*)d_in[0];   // (32,1024,32) f32
    const float* emb    = (const float*)d_in[1];   // (8192,32)   f32
    float* out = (float*)d_out;

    // d_out layout (return-order flat): z_q | indices(as float) | q_loss | e_loss
    float* zq_out   = out;
    float* idx_out  = out + (size_t)N_TOK * E_DIM;
    float* loss_out = idx_out + N_TOK;

    // workspace layout
    char* ws = (char*)d_ws;
    float* emb_norm  = (float*)ws;                                    // 1 MB
    float* flat_norm = (float*)(ws + (size_t)N_EMB * E_DIM * 4);      // 4 MB
    int*   bidx      = (int*)  (ws + (size_t)(N_EMB + N_TOK) * E_DIM * 4);
    float* partials  = (float*)(ws + (size_t)(N_EMB + N_TOK) * E_DIM * 4 + (size_t)N_TOK * 4);

    row_l2norm_kernel<<<N_EMB / 256, 256, 0, stream>>>(emb, emb_norm, N_EMB);
    row_l2norm_kernel<<<N_TOK / 256, 256, 0, stream>>>(hidden, flat_norm, N_TOK);

    vq_search_kernel<<<N_TOK / 256, 256, 0, stream>>>(flat_norm, emb_norm, bidx, idx_out);

    gather_loss_kernel<<<N_TOK / 256, 256, 0, stream>>>(emb, flat_norm, bidx, zq_out, partials);
    finalize_loss_kernel<<<1, 64, 0, stream>>>(partials, N_TOK / 256, loss_out);
}